// GNNEncoder_62972810494478
// MI455X (gfx1250) — compile-verified
//
#include <hip/hip_runtime.h>
#include <hip/hip_bf16.h>
#include <cstdint>
#include <cstddef>

typedef __attribute__((ext_vector_type(2))) float v2f;
typedef __attribute__((ext_vector_type(8))) float v8f;

#define NSTUD 50000
#define NITEM 20000
#define EMBD  64
#define ROWP  68       // padded LDS row stride (floats) -> conflict-free column walks
#define MHALF 2080     // 16 kpairs * 128 words + 32-word skew (bank-offsets the odd half)

// ---------------------------------------------------------------------------
// Precompute kernel: collapse attention matmuls to vectors.
//   vvec = W_rel @ lg_W[64:128]; u1 = Wns_W @ Was[:64]; u2 = Wncs_W @ Was[64:]
//   u3 = Wni_W @ Wai[:64]; u4 = Wnci_W @ Wai[64:]; c_s, c_i = collapsed biases
// CV layout: [0:64) vvec | [64:128) lg_b | [128:192) u1 | [192:256) u2
//            [256:320) u3 | [320:384) u4 | [384] c_s | [385] c_i
// ---------------------------------------------------------------------------
__global__ void gnn_pre(const float* __restrict__ W_rel,
                        const float* __restrict__ lg_W,
                        const float* __restrict__ lg_b,
                        const float* __restrict__ Wns_W,  const float* __restrict__ Wns_b,
                        const float* __restrict__ Wncs_W, const float* __restrict__ Wncs_b,
                        const float* __restrict__ Was_W,  const float* __restrict__ Was_b,
                        const float* __restrict__ Wni_W,  const float* __restrict__ Wni_b,
                        const float* __restrict__ Wnci_W, const float* __restrict__ Wnci_b,
                        const float* __restrict__ Wai_W,  const float* __restrict__ Wai_b,
                        float* __restrict__ CV)
{
    const int j = threadIdx.x;   // 64 threads
    if (j >= EMBD) return;
    float vv = 0.f, u1 = 0.f, u2 = 0.f, u3 = 0.f, u4 = 0.f;
    for (int k = 0; k < EMBD; ++k) {
        vv += W_rel[k]             * lg_W[(EMBD + k) * EMBD + j];
        u1 += Wns_W [j * EMBD + k] * Was_W[k];
        u2 += Wncs_W[j * EMBD + k] * Was_W[EMBD + k];
        u3 += Wni_W [j * EMBD + k] * Wai_W[k];
        u4 += Wnci_W[j * EMBD + k] * Wai_W[EMBD + k];
    }
    CV[j]           = vv;
    CV[EMBD + j]    = lg_b[j];
    CV[2*EMBD + j]  = u1;
    CV[3*EMBD + j]  = u2;
    CV[4*EMBD + j]  = u3;
    CV[5*EMBD + j]  = u4;
    if (j == 0) {
        float cs = Was_b[0], ci = Wai_b[0];
        for (int k = 0; k < EMBD; ++k) {
            cs += Wns_b[k] * Was_W[k] + Wncs_b[k] * Was_W[EMBD + k];
            ci += Wni_b[k] * Wai_W[k] + Wnci_b[k] * Wai_W[EMBD + k];
        }
        CV[6*EMBD + 0] = cs;
        CV[6*EMBD + 1] = ci;
    }
}

__global__ void gnn_zero(float* __restrict__ p, int n)
{
    int i = blockIdx.x * blockDim.x + threadIdx.x;
    if (i < n) p[i] = 0.f;
}

// ---------------------------------------------------------------------------
// Per-edge kernel. One wave = 16 edges. Block = 128 threads = 4 waves.
// PASS 1: compute exp(attn) for both sides, atomicAdd into softmax denoms.
// PASS 2: recompute cs tiles, alpha = exp/denom, atomicAdd alpha*cs into out.
// Matvec e @ lg_W[:64] done as 16x64 @ 64x64 via V_WMMA_F32_16X16X4_F32.
// B matrix staged as K-pair-packed float2 (even/odd kpairs in bank-skewed
// halves) so each B operand is one conflict-free ds_load_b64.
// ---------------------------------------------------------------------------
template <int PASS>
__global__ __launch_bounds__(128)
void gnn_edge(const int*   __restrict__ e_s,
              const int*   __restrict__ e_i,
              const float* __restrict__ resp,
              const float* __restrict__ student_emb,
              const float* __restrict__ item_emb,
              const float* __restrict__ lg_W,
              const float* __restrict__ CV,
              float* __restrict__ denom_s, float* __restrict__ denom_i,
              float* __restrict__ exp_s,   float* __restrict__ exp_i,
              float* __restrict__ out_s,   float* __restrict__ out_i,
              int E)
{
    // Msh: kpair-packed lg_W[:64]. word(kp,n,j) = (kp&1)*MHALF + (kp>>1)*128 + 2n + j
    // holds lg_W[(2*kp+j)*64 + n]; the odd-kpair half is skewed by 32 words so the
    // two wave halves of a b64 read cover disjoint bank windows.
    __shared__ float Msh[2 * MHALF];
    __shared__ float CVl[6 * EMBD + 2];
    __shared__ float EQ [4][16 * ROWP];
    __shared__ float ESX[4][16 * ROWP];
    __shared__ float CSQ[4][16 * ROWP];
    __shared__ float CSS[4][16 * ROWP];
    __shared__ float RW [4][16];

    const int tid  = threadIdx.x;
    const int wave = tid >> 5;
    const int lane = tid & 31;
    const int row  = lane & 15;   // A-row / B-col owned by this lane
    const int kh   = lane >> 4;   // which K-pair half (WMMA f32 A/B layout)

    // Block prologue: stage packed matrix + collapsed vectors
    for (int i = tid; i < EMBD * EMBD; i += 128) {
        const int kp = i >> 7;          // kpair 0..31
        const int n  = (i >> 1) & 63;   // column
        const int j  = i & 1;           // which row of the pair
        Msh[(kp & 1) * MHALF + (kp >> 1) * 128 + 2 * n + j] =
            lg_W[(2 * kp + j) * EMBD + n];
    }
    for (int i = tid; i < 6 * EMBD + 2; i += 128) CVl[i] = CV[i];

    float* EQw  = EQ [wave];
    float* ESw  = ESX[wave];
    float* CSQw = CSQ[wave];
    float* CSSw = CSS[wave];
    float* RWw  = RW [wave];

    const int  ebase  = (blockIdx.x * 4 + wave) * 16;
    const bool active = (ebase < E);           // wave-uniform

    int myS = 0, myI = 0;
    if (active) {
        int e = ebase + row;
        if (e >= E) e = E - 1;                 // clamp (writes are masked later)
        myS = e_s[e];
        myI = e_i[e];
        const float* eqp = item_emb    + (size_t)myI * EMBD + kh * 32;
        const float* esp = student_emb + (size_t)myS * EMBD + kh * 32;
        #pragma unroll
        for (int t = 0; t < 8; ++t) {
            float4 vq = *(const float4*)(eqp + 4 * t);
            float4 vs = *(const float4*)(esp + 4 * t);
            *(float4*)&EQw[row * ROWP + kh * 32 + 4 * t] = vq;
            *(float4*)&ESw[row * ROWP + kh * 32 + 4 * t] = vs;
        }
        if (kh == 0) RWw[row] = resp[e];
    }
    __syncthreads();

    // Attention gather-side dots (pass 1 only): d1 = es.u1, d3 = eq.u3
    float d1 = 0.f, d3 = 0.f;
    if (active && PASS == 1) {
        #pragma unroll 8
        for (int t = 0; t < 32; ++t) {
            const int c = kh * 32 + t;
            d1 += ESw[row * ROWP + c] * CVl[2*EMBD + c];
            d3 += EQw[row * ROWP + c] * CVl[4*EMBD + c];
        }
        d1 += __shfl_xor(d1, 16, 32);
        d3 += __shfl_xor(d3, 16, 32);
    }

    if (active) {
        // ---- WMMA: [16x64] @ [64x64] for both eq and es tiles ----
        const float* Mp = Msh + kh * MHALF;    // per-lane half select (kpair parity)

        v8f accq[4], accs[4];
        #pragma unroll
        for (int nt = 0; nt < 4; ++nt)
            #pragma unroll
            for (int j = 0; j < 8; ++j) { accq[nt][j] = 0.f; accs[nt][j] = 0.f; }

        #pragma unroll
        for (int k = 0; k < 16; ++k) {
            const int ac = 4 * k + 2 * kh;     // K column pair for this lane
            v2f aQ, aS;
            aQ[0] = EQw[row * ROWP + ac];  aQ[1] = EQw[row * ROWP + ac + 1];
            aS[0] = ESw[row * ROWP + ac];  aS[1] = ESw[row * ROWP + ac + 1];
            #pragma unroll
            for (int nt = 0; nt < 4; ++nt) {
                const v2f b = *(const v2f*)&Mp[k * 128 + 2 * (nt * 16 + row)];
                accq[nt] = __builtin_amdgcn_wmma_f32_16x16x4_f32(
                    false, aQ, false, b, (short)0, accq[nt], false, false);
                accs[nt] = __builtin_amdgcn_wmma_f32_16x16x4_f32(
                    false, aS, false, b, (short)0, accs[nt], false, false);
            }
        }

        // ---- ls = acc + r*vvec + lg_b ; cs = emb * ls ; spill cs to LDS ----
        #pragma unroll
        for (int nt = 0; nt < 4; ++nt) {
            const int   n  = nt * 16 + row;
            const float vv = CVl[n];
            const float lb = CVl[EMBD + n];
            #pragma unroll
            for (int r = 0; r < 8; ++r) {
                const int   m    = r + 8 * kh;       // C-layout row for VGPR r
                const float rm   = RWw[m];
                const float base = rm * vv + lb;
                const float lsq  = accq[nt][r] + base;
                const float lss  = accs[nt][r] + base;
                CSQw[m * ROWP + n] = EQw[m * ROWP + n] * lsq;
                CSSw[m * ROWP + n] = ESw[m * ROWP + n] * lss;
            }
        }
    }
    __syncthreads();

    if (active) {
        const int  e     = ebase + row;
        const bool valid = (e < E);

        if (PASS == 1) {
            // attn dots on cs tiles, then exp + denominator scatter
            float dq = 0.f, dcs = 0.f;
            #pragma unroll 8
            for (int t = 0; t < 32; ++t) {
                const int c = kh * 32 + t;
                dq  += CSQw[row * ROWP + c] * CVl[3*EMBD + c];   // u2
                dcs += CSSw[row * ROWP + c] * CVl[5*EMBD + c];   // u4
            }
            dq  += __shfl_xor(dq, 16, 32);
            dcs += __shfl_xor(dcs, 16, 32);

            float a_s = d1 + dq  + CVl[6*EMBD + 0];
            float a_i = d3 + dcs + CVl[6*EMBD + 1];
            a_s = a_s > 0.f ? a_s : 0.01f * a_s;   // leaky_relu(0.01)
            a_i = a_i > 0.f ? a_i : 0.01f * a_i;
            const float Es = __expf(a_s);
            const float Ei = __expf(a_i);
            if (valid && kh == 0) {
                exp_s[e] = Es;
                exp_i[e] = Ei;
                atomicAdd(&denom_s[myS], Es);
                atomicAdd(&denom_i[myI], Ei);
            }
        } else {
            // alpha-weighted scatter of cs tiles into the outputs
            if (valid) {
                const float alS = exp_s[e] / denom_s[myS];
                const float alI = exp_i[e] / denom_i[myI];
                float* po = out_s + (size_t)myS * EMBD;
                float* pi = out_i + (size_t)myI * EMBD;
                #pragma unroll 4
                for (int t = 0; t < 32; ++t) {
                    const int c = kh * 32 + t;
                    atomicAdd(&po[c], alS * CSQw[row * ROWP + c]);
                    atomicAdd(&pi[c], alI * CSSw[row * ROWP + c]);
                }
            }
        }
    }
}

// ---------------------------------------------------------------------------
extern "C" void kernel_launch(void* const* d_in, const int* in_sizes, int n_in,
                              void* d_out, int out_size, void* d_ws, size_t ws_size,
                              hipStream_t stream)
{
    const int E = in_sizes[1];                       // responses: [E]

    const int*   edge  = (const int*)  d_in[0];      // [2,E]
    const int*   e_s   = edge;
    const int*   e_i   = edge + E;
    const float* resp  = (const float*)d_in[1];
    const float* semb  = (const float*)d_in[2];
    const float* iemb  = (const float*)d_in[3];
    const float* W_rel = (const float*)d_in[4];
    const float* lg_W  = (const float*)d_in[5];
    const float* lg_b  = (const float*)d_in[6];
    const float* WnsW  = (const float*)d_in[7];
    const float* Wnsb  = (const float*)d_in[8];
    const float* WncsW = (const float*)d_in[9];
    const float* Wncsb = (const float*)d_in[10];
    const float* WasW  = (const float*)d_in[11];
    const float* Wasb  = (const float*)d_in[12];
    const float* WniW  = (const float*)d_in[13];
    const float* Wnib  = (const float*)d_in[14];
    const float* WnciW = (const float*)d_in[15];
    const float* Wncib = (const float*)d_in[16];
    const float* WaiW  = (const float*)d_in[17];
    const float* Waib  = (const float*)d_in[18];

    // workspace (floats): denom_s | denom_i | exp_s[E] | exp_i[E] | CV[386]
    float* ws      = (float*)d_ws;
    float* denom_s = ws;
    float* denom_i = ws + NSTUD;
    float* exp_s   = ws + NSTUD + NITEM;
    float* exp_i   = exp_s + E;
    float* CV      = exp_i + E;

    float* out   = (float*)d_out;
    float* out_s = out;
    float* out_i = out + (size_t)NSTUD * EMBD;

    // 1) collapse attention weights into vectors/scalars
    gnn_pre<<<1, 64, 0, stream>>>(W_rel, lg_W, lg_b,
                                  WnsW, Wnsb, WncsW, Wncsb, WasW, Wasb,
                                  WniW, Wnib, WnciW, Wncib, WaiW, Waib, CV);

    // 2) zero softmax denominators
    const int zn = NSTUD + NITEM;
    gnn_zero<<<(zn + 255) / 256, 256, 0, stream>>>(ws, zn);

    // 3) out = [student_emb ; item_emb]  (scatter-adds accumulate on top)
    hipMemcpyAsync(out_s, semb, (size_t)NSTUD * EMBD * sizeof(float),
                   hipMemcpyDeviceToDevice, stream);
    hipMemcpyAsync(out_i, iemb, (size_t)NITEM * EMBD * sizeof(float),
                   hipMemcpyDeviceToDevice, stream);

    // 4) pass 1: exp(attn) + denominators;  5) pass 2: weighted aggregation
    const int nb = (E + 63) / 64;                    // 4 waves x 16 edges
    gnn_edge<1><<<nb, 128, 0, stream>>>(e_s, e_i, resp, semb, iemb, lg_W, CV,
                                        denom_s, denom_i, exp_s, exp_i,
                                        out_s, out_i, E);
    gnn_edge<2><<<nb, 128, 0, stream>>>(e_s, e_i, resp, semb, iemb, lg_W, CV,
                                        denom_s, denom_i, exp_s, exp_i,
                                        out_s, out_i, E);
}